// AdaptiveBoxBlur2d_74311524155669
// MI455X (gfx1250) — compile-verified
//
#include <hip/hip_runtime.h>
#include <math.h>

#define NN 2
#define CC 8
#define HH 1024
#define WW 1024
#define HWSZ (HH * WW)
#define NC (NN * CC)
#define EPSF 1e-5f

typedef float v2f __attribute__((ext_vector_type(2)));
typedef float v8f __attribute__((ext_vector_type(8)));

// ---------------------------------------------------------------------------
// Kernel 1: per-(n,c) mean and unbiased std.
// ---------------------------------------------------------------------------
__global__ void abb_stats_kernel(const float* __restrict__ x, float* __restrict__ stats) {
  __shared__ double s_sum[256];
  __shared__ double s_sq[256];
  const int nc = blockIdx.x;
  const float* p = x + (size_t)nc * HWSZ;
  double s = 0.0, q = 0.0;
  for (int i = threadIdx.x; i < HWSZ; i += 256) {
    float v = p[i];
    s += (double)v;
    q += (double)v * (double)v;
  }
  s_sum[threadIdx.x] = s;
  s_sq[threadIdx.x] = q;
  __syncthreads();
  for (int off = 128; off > 0; off >>= 1) {
    if (threadIdx.x < off) {
      s_sum[threadIdx.x] += s_sum[threadIdx.x + off];
      s_sq[threadIdx.x] += s_sq[threadIdx.x + off];
    }
    __syncthreads();
  }
  if (threadIdx.x == 0) {
    double sum = s_sum[0], sq = s_sq[0];
    double mean = sum / (double)HWSZ;
    double var = (sq - sum * sum / (double)HWSZ) / (double)(HWSZ - 1);
    if (var < 0.0) var = 0.0;
    stats[nc] = (float)mean;           // mean
    stats[NC + nc] = (float)sqrt(var); // unbiased std
  }
}

// ---------------------------------------------------------------------------
// Kernel 2: row-wise (axis=-1) prefix sum of (x-mean)/(std+eps) via WMMA.
// One wave owns a 16-row x 1024-col band. Per 16x16 tile: Y = X * U + Carry,
// U = upper-triangular ones, done as 4 chained V_WMMA_F32_16X16X4_F32.
// ---------------------------------------------------------------------------
__global__ void abb_row_scan_kernel(const float* __restrict__ x,
                                    const float* __restrict__ stats,
                                    float* __restrict__ I) {
  __shared__ float lds[8][16 * 17];
  const int wave = threadIdx.x >> 5;
  const int lane = threadIdx.x & 31;
  const int band = blockIdx.x * 8 + wave; // 0..1023
  const int nc = band >> 6;               // / (HH/16)
  const int r0 = (band & 63) << 4;
  const float mean = stats[nc];
  const float istd = 1.0f / (stats[NC + nc] + EPSF);
  const float* src = x + (size_t)nc * HWSZ;
  float* dst = I + (size_t)nc * HWSZ;
  const int sub = lane >> 4; // 0/1 half-wave
  const int col = lane & 15; // N index (and tile column for staging)
  float* L = lds[wave];

  // Carry accumulator in C/D layout: acc[v] = carry for row (v + 8*sub).
  v8f acc = {};
  for (int t = 0; t < WW / 16; ++t) {
    const int c0 = t * 16;
    // Stage normalized 16x16 tile into LDS (coalesced 64B row chunks).
#pragma unroll
    for (int i = 0; i < 8; ++i) {
      const int r = i * 2 + sub;
      float v = src[(size_t)(r0 + r) * WW + (c0 + col)];
      L[r * 17 + col] = (v - mean) * istd;
    }
    __syncthreads();

    v8f d = acc;
#pragma unroll
    for (int p = 0; p < 4; ++p) {
      const int k0 = p * 4 + 2 * sub; // K index held by this lane-half
      v2f a, b;
      // A = X slice (16x4): A[m][k], m = lane&15.
      a.x = L[col * 17 + k0];
      a.y = L[col * 17 + k0 + 1];
      // B = U slice (4x16): U[k][n] = (k <= n).
      b.x = (k0 <= col) ? 1.0f : 0.0f;
      b.y = (k0 + 1 <= col) ? 1.0f : 0.0f;
      d = __builtin_amdgcn_wmma_f32_16x16x4_f32(false, a, false, b, (short)0, d,
                                                false, false);
    }
    // Write Y tile (row = v + 8*sub, col = lane&15).
#pragma unroll
    for (int v = 0; v < 8; ++v) {
      dst[(size_t)(r0 + v + 8 * sub) * WW + (c0 + col)] = d[v];
    }
    // New carry per row = Y[:,15]: lives in lane 15 (rows 0-7) / lane 31 (rows 8-15).
#pragma unroll
    for (int v = 0; v < 8; ++v) acc[v] = __shfl(d[v], 15, 16);
    __syncthreads();
  }
}

// ---------------------------------------------------------------------------
// Kernel 3: column-wise (axis=-2) prefix sum of I, in place, via WMMA.
// One wave owns a 1024-row x 16-col band. Per tile: Y = Lmat * X + Carry.
// ---------------------------------------------------------------------------
__global__ void abb_col_scan_kernel(float* __restrict__ I) {
  __shared__ float lds[8][16 * 17];
  const int wave = threadIdx.x >> 5;
  const int lane = threadIdx.x & 31;
  const int band = blockIdx.x * 8 + wave; // 0..1023
  const int nc = band >> 6;
  const int c0 = (band & 63) << 4;
  float* plane = I + (size_t)nc * HWSZ;
  const int sub = lane >> 4;
  const int col = lane & 15;
  float* L = lds[wave];

  float carry = 0.0f; // per-column carry for column (lane&15)
  for (int t = 0; t < HH / 16; ++t) {
    const int r0 = t * 16;
#pragma unroll
    for (int i = 0; i < 8; ++i) {
      const int r = i * 2 + sub;
      L[r * 17 + col] = plane[(size_t)(r0 + r) * WW + (c0 + col)];
    }
    __syncthreads();

    v8f d;
#pragma unroll
    for (int v = 0; v < 8; ++v) d[v] = carry; // C[i][j] = carry[j]
#pragma unroll
    for (int p = 0; p < 4; ++p) {
      const int k0 = p * 4 + 2 * sub;
      v2f a, b;
      // A = lower-triangular ones slice: A[m][k] = (k <= m), m = lane&15.
      a.x = (k0 <= col) ? 1.0f : 0.0f;
      a.y = (k0 + 1 <= col) ? 1.0f : 0.0f;
      // B = X slice (4x16): row k0(+1) of the staged tile.
      b.x = L[k0 * 17 + col];
      b.y = L[(k0 + 1) * 17 + col];
      d = __builtin_amdgcn_wmma_f32_16x16x4_f32(false, a, false, b, (short)0, d,
                                                false, false);
    }
#pragma unroll
    for (int v = 0; v < 8; ++v) {
      plane[(size_t)(r0 + v + 8 * sub) * WW + (c0 + col)] = d[v];
    }
    // New carry per column n = Y[15][n] = D VGPR7, lane 16+n.
    carry = __shfl(d[7], 16 + col, 32);
    __syncthreads();
  }
}

// ---------------------------------------------------------------------------
// Kernel 4: 4-corner bilinear gather (grid_sample border, align_corners) +
// rescale. One thread per (n,h,w), loops over C=8 channels.
// ---------------------------------------------------------------------------
__device__ __forceinline__ void abb_coord(float g, int S, int& i0, int& i1, float& f) {
  float i = (g + 1.0f) * 0.5f * (float)(S - 1);
  i = fminf(fmaxf(i, 0.0f), (float)(S - 1));
  float fl = floorf(i);
  f = i - fl;
  i0 = (int)fl;
  i1 = (i0 + 1 < S) ? (i0 + 1) : (S - 1);
}

__global__ void abb_gather_kernel(const float* __restrict__ ks,
                                  const float* __restrict__ I,
                                  const float* __restrict__ stats,
                                  float* __restrict__ out) {
  const int idx = blockIdx.x * 256 + threadIdx.x; // over N*H*W
  if (idx >= NN * HH * WW) return;
  const int w = idx % WW;
  const int h = (idx / WW) % HH;
  const int n = idx / (HH * WW);

  const float kx = ks[(size_t)idx * 2 + 0]; // pixels, [1,16)
  const float ky = ks[(size_t)idx * 2 + 1];
  const float gxv = -1.0f + 2.0f * (float)w / (float)(WW - 1) - 1.0f / (float)WW;
  const float gyv = -1.0f + 2.0f * (float)h / (float)(HH - 1) - 1.0f / (float)HH;
  const float kxn = kx / (float)WW;
  const float kyn = ky / (float)HH;

  int xp0, xp1, xm0, xm1, yp0, yp1, ym0, ym1;
  float fxp, fxm, fyp, fym;
  abb_coord(gxv + kxn, WW, xp0, xp1, fxp);
  abb_coord(gxv - kxn, WW, xm0, xm1, fxm);
  abb_coord(gyv + kyn, HH, yp0, yp1, fyp);
  abb_coord(gyv - kyn, HH, ym0, ym1, fym);

  const float inv_area = 1.0f / (kx * ky + EPSF);

#pragma unroll
  for (int c = 0; c < CC; ++c) {
    const float* P = I + (size_t)(n * CC + c) * HWSZ;
    // bilinear taps
    float app = P[(size_t)yp0 * WW + xp0] * (1.0f - fxp) * (1.0f - fyp)
              + P[(size_t)yp0 * WW + xp1] * fxp * (1.0f - fyp)
              + P[(size_t)yp1 * WW + xp0] * (1.0f - fxp) * fyp
              + P[(size_t)yp1 * WW + xp1] * fxp * fyp;            // (+x,+y) = A
    float apm = P[(size_t)ym0 * WW + xp0] * (1.0f - fxp) * (1.0f - fym)
              + P[(size_t)ym0 * WW + xp1] * fxp * (1.0f - fym)
              + P[(size_t)ym1 * WW + xp0] * (1.0f - fxp) * fym
              + P[(size_t)ym1 * WW + xp1] * fxp * fym;            // (+x,-y) = B
    float amp = P[(size_t)yp0 * WW + xm0] * (1.0f - fxm) * (1.0f - fyp)
              + P[(size_t)yp0 * WW + xm1] * fxm * (1.0f - fyp)
              + P[(size_t)yp1 * WW + xm0] * (1.0f - fxm) * fyp
              + P[(size_t)yp1 * WW + xm1] * fxm * fyp;            // (-x,+y) = C
    float amm = P[(size_t)ym0 * WW + xm0] * (1.0f - fxm) * (1.0f - fym)
              + P[(size_t)ym0 * WW + xm1] * fxm * (1.0f - fym)
              + P[(size_t)ym1 * WW + xm0] * (1.0f - fxm) * fym
              + P[(size_t)ym1 * WW + xm1] * fxm * fym;            // (-x,-y) = D

    float res = app - apm - amp + amm;
    float mean = stats[n * CC + c];
    float stdv = stats[NC + n * CC + c];
    out[(size_t)(n * CC + c) * HWSZ + (size_t)h * WW + w] =
        res * inv_area * stdv + mean;
  }
}

// ---------------------------------------------------------------------------
extern "C" void kernel_launch(void* const* d_in, const int* in_sizes, int n_in,
                              void* d_out, int out_size, void* d_ws, size_t ws_size,
                              hipStream_t stream) {
  (void)in_sizes; (void)n_in; (void)out_size; (void)ws_size;
  const float* x = (const float*)d_in[0];
  const float* ks = (const float*)d_in[1];
  float* out = (float*)d_out;
  float* stats = (float*)d_ws;        // 32 floats (mean[16], std[16])
  float* I = (float*)d_ws + 64;       // integral image, 67 MB, 256B aligned

  abb_stats_kernel<<<NC, 256, 0, stream>>>(x, stats);
  abb_row_scan_kernel<<<128, 256, 0, stream>>>(x, stats, I);
  abb_col_scan_kernel<<<128, 256, 0, stream>>>(I);
  abb_gather_kernel<<<(NN * HH * WW + 255) / 256, 256, 0, stream>>>(ks, I, stats, out);
}